// IRNetLayer_24678882083160
// MI455X (gfx1250) — compile-verified
//
#include <hip/hip_runtime.h>
#include <math.h>

typedef __attribute__((ext_vector_type(2))) float v2f;
typedef __attribute__((ext_vector_type(8))) float v8f;

#define N_NODES 20000
#define NDIM    256
#define NHEADS  8
#define DK      32
#define DEG     16

// ---------------------------------------------------------------------------
// fp32 WMMA: D = A(16x4) * B(4x16) + C(16x16), all f32.
// 8-arg pattern: (neg_a, A, neg_b, B, c_mod, C, reuse_a, reuse_b)
// ---------------------------------------------------------------------------
__device__ __forceinline__ v8f wmma_f32_16x16x4(v2f a, v2f b, v8f c) {
  return __builtin_amdgcn_wmma_f32_16x16x4_f32(
      false, a, false, b, (short)0, c, false, false);
}

// ---------------------------------------------------------------------------
// Generic WMMA GEMM:  C[M,N] = A[M,K] @ B[K,N] (+bias) (+ReLU) (+residual)
// A row-major, B row-major. One wave computes a 32x64 tile: two 16-row
// subtiles share the same four B fragments -> 8 WMMAs per 10 loads per K-step
// (vs 4 WMMAs per 9 loads with a 16x64 tile), easing VMEM issue pressure.
// M % 32 == 0, N % 64 == 0, K % 4 == 0 guaranteed by caller.
//
// Fragment layouts (CDNA5 ISA 7.12.2, f32):
//   A 16x4 : lanes 0-15 -> M=lane, VGPR0=K, VGPR1=K+1 ; lanes 16-31 -> K+2,K+3
//   B 4x16 : lanes 0-15 -> N=lane, VGPR0=K row, VGPR1=K+1 ; lanes 16-31 -> K+2,K+3
//   C 16x16: VGPR r: lanes 0-15 -> (M=r, N=lane); lanes 16-31 -> (M=r+8, N=lane-16)
// ---------------------------------------------------------------------------
template <int RELU>
__global__ __launch_bounds__(256) void gemm_wmma_kernel(
    const float* __restrict__ A, const float* __restrict__ B,
    const float* __restrict__ bias, const float* __restrict__ res,
    float* __restrict__ C, int M, int N, int K) {
  const int lane = threadIdx.x & 31;
  const int wave = threadIdx.x >> 5;
  const int nt64 = N >> 6;                 // 64-wide column tiles
  const int T    = (M >> 5) * nt64;        // total wave-tiles (32-row tiles)
  const int wt   = blockIdx.x * 8 + wave;  // wave-uniform -> EXEC stays all-1s
  if (wt >= T) return;

  const int row0 = (wt / nt64) << 5;
  const int col0 = (wt % nt64) << 6;
  const int half = lane >> 4;   // 0: lanes 0-15, 1: lanes 16-31
  const int l16  = lane & 15;

  // acc[s][t]: row-subtile s (rows row0+16s..), col-subtile t (cols col0+16t..)
  v8f acc[2][4];
#pragma unroll
  for (int s = 0; s < 2; ++s)
#pragma unroll
    for (int t = 0; t < 4; ++t) acc[s][t] = (v8f){};

  const float* Arow0 = A + (size_t)(row0 + l16) * K + (half << 1);
  const float* Arow1 = Arow0 + (size_t)16 * K;

  for (int k = 0; k < K; k += 4) {
    v2f a0, a1;
    a0.x = Arow0[k]; a0.y = Arow0[k + 1];
    a1.x = Arow1[k]; a1.y = Arow1[k + 1];

    const float* b0 = B + (size_t)(k + (half << 1)) * N + col0 + l16;
    const float* b1 = b0 + N;
    v2f bf[4];
#pragma unroll
    for (int t = 0; t < 4; ++t) {
      bf[t].x = b0[t * 16];
      bf[t].y = b1[t * 16];
    }

#pragma unroll
    for (int t = 0; t < 4; ++t) {
      acc[0][t] = wmma_f32_16x16x4(a0, bf[t], acc[0][t]);
      acc[1][t] = wmma_f32_16x16x4(a1, bf[t], acc[1][t]);
    }
  }

#pragma unroll
  for (int s = 0; s < 2; ++s) {
#pragma unroll
    for (int r = 0; r < 8; ++r) {
      const int m = row0 + s * 16 + r + (half << 3);
      const size_t base = (size_t)m * N + col0 + l16;
#pragma unroll
      for (int t = 0; t < 4; ++t) {
        float v = acc[s][t][r];
        if (bias) v += bias[col0 + l16 + t * 16];
        if (RELU) v = fmaxf(v, 0.0f);
        if (res)  v += res[base + (size_t)(t * 16)];
        C[base + (size_t)(t * 16)] = v;
      }
    }
  }
}

// ---------------------------------------------------------------------------
// Edge attention + aggregation. dst = repeat(arange(N), DEG) (sorted), so node
// n owns edges [n*DEG, (n+1)*DEG). Block = node, wave = head, lane = D_K dim
// (D_K == 32 == wave32). score = <k[src], q[n]> via shfl_xor reduction.
// All gathered rows live in the 192MB L2 (q/k/v total 61MB), so the 655MB of
// gather traffic is cache-resident, not HBM.
// ---------------------------------------------------------------------------
__global__ __launch_bounds__(256) void edge_attn_kernel(
    const float* __restrict__ q, const float* __restrict__ k,
    const float* __restrict__ v, const int* __restrict__ src,
    float* __restrict__ o) {
  const int node = blockIdx.x;
  const int head = threadIdx.x >> 5;
  const int lane = threadIdx.x & 31;
  const int d    = head * DK + lane;

  const float ql = q[(size_t)node * NDIM + d];
  const float inv_sqrt_dk = 0.17677669529663687f;  // 1/sqrt(32)

  float acc = 0.0f, z = 0.0f;
#pragma unroll
  for (int e = 0; e < DEG; ++e) {
    const int s = src[node * DEG + e];
    float p = ql * k[(size_t)s * NDIM + d];
#pragma unroll
    for (int off = 16; off > 0; off >>= 1) p += __shfl_xor(p, off, 32);
    float sc = p * inv_sqrt_dk;
    sc = fminf(fmaxf(sc, -10.0f), 10.0f);
    const float w = expf(sc);
    acc += w * v[(size_t)s * NDIM + d];
    z += w;
  }
  o[(size_t)node * NDIM + d] = acc / z;
}

// ---------------------------------------------------------------------------
// Row LayerNorm over NDIM=256, one block per row. Safe for in == out.
// ---------------------------------------------------------------------------
__global__ __launch_bounds__(256) void layernorm_kernel(
    const float* in, const float* __restrict__ g, const float* __restrict__ b,
    float* out) {
  __shared__ float red[NDIM];
  const int row = blockIdx.x;
  const int tid = threadIdx.x;

  const float x = in[(size_t)row * NDIM + tid];
  red[tid] = x;
  __syncthreads();
  for (int s = NDIM / 2; s > 0; s >>= 1) {
    if (tid < s) red[tid] += red[tid + s];
    __syncthreads();
  }
  const float mu = red[0] * (1.0f / NDIM);
  __syncthreads();

  const float dlt = x - mu;
  red[tid] = dlt * dlt;
  __syncthreads();
  for (int s = NDIM / 2; s > 0; s >>= 1) {
    if (tid < s) red[tid] += red[tid + s];
    __syncthreads();
  }
  const float var = red[0] * (1.0f / NDIM);
  const float r = rsqrtf(var + 1e-5f);
  out[(size_t)row * NDIM + tid] = dlt * r * g[tid] + b[tid];
}

// ---------------------------------------------------------------------------
// Host-side orchestration.
// Inputs: 0 x, 1 src, 2 dst(unused, sorted-by-construction), 3 Wq, 4 bq,
//         5 Wk, 6 Wv, 7 Wo, 8 bo, 9 ln1_g, 10 ln1_b, 11 W1, 12 b1,
//         13 W2, 14 b2, 15 ln2_g, 16 ln2_b
// ---------------------------------------------------------------------------
extern "C" void kernel_launch(void* const* d_in, const int* in_sizes, int n_in,
                              void* d_out, int out_size, void* d_ws,
                              size_t ws_size, hipStream_t stream) {
  const float* x    = (const float*)d_in[0];
  const int*   src  = (const int*)d_in[1];
  const float* Wq   = (const float*)d_in[3];
  const float* bq   = (const float*)d_in[4];
  const float* Wk   = (const float*)d_in[5];
  const float* Wv   = (const float*)d_in[6];
  const float* Wo   = (const float*)d_in[7];
  const float* bo   = (const float*)d_in[8];
  const float* ln1g = (const float*)d_in[9];
  const float* ln1b = (const float*)d_in[10];
  const float* W1   = (const float*)d_in[11];
  const float* b1   = (const float*)d_in[12];
  const float* W2   = (const float*)d_in[13];
  const float* b2   = (const float*)d_in[14];
  const float* ln2g = (const float*)d_in[15];
  const float* ln2b = (const float*)d_in[16];
  float* out = (float*)d_out;

  const int N = N_NODES;
  const size_t S = (size_t)N * NDIM;  // elements per node-feature buffer

  float* q  = (float*)d_ws;  // [0S,1S)
  float* kk = q + S;         // [1S,2S)
  float* vv = kk + S;        // [2S,3S)
  float* o  = vv + S;        // [3S,4S)
  float* h  = o + S;         // [4S,5S)
  float* a  = q;             // FFN mid [0S,4S): reuses q/k/v/o after attention

  auto blocks = [](int M, int Ncols) {
    int T = (M / 32) * (Ncols / 64);
    return (T + 7) / 8;
  };

  // q = x@Wq + bq ; k = x@Wk ; v = x@Wv
  gemm_wmma_kernel<0><<<blocks(N, NDIM), 256, 0, stream>>>(x, Wq, bq, nullptr, q,  N, NDIM, NDIM);
  gemm_wmma_kernel<0><<<blocks(N, NDIM), 256, 0, stream>>>(x, Wk, nullptr, nullptr, kk, N, NDIM, NDIM);
  gemm_wmma_kernel<0><<<blocks(N, NDIM), 256, 0, stream>>>(x, Wv, nullptr, nullptr, vv, N, NDIM, NDIM);

  // edge attention + per-node aggregation -> o
  edge_attn_kernel<<<N, 256, 0, stream>>>(q, kk, vv, src, o);

  // h = LN1(x + o@Wo + bo)
  gemm_wmma_kernel<0><<<blocks(N, NDIM), 256, 0, stream>>>(o, Wo, bo, x, h, N, NDIM, NDIM);
  layernorm_kernel<<<N, 256, 0, stream>>>(h, ln1g, ln1b, h);

  // out = LN2(h + relu(h@W1 + b1)@W2 + b2)
  gemm_wmma_kernel<1><<<blocks(N, 4 * NDIM), 256, 0, stream>>>(h, W1, b1, nullptr, a, N, 4 * NDIM, NDIM);
  gemm_wmma_kernel<0><<<blocks(N, NDIM), 256, 0, stream>>>(a, W2, b2, h, out, N, NDIM, 4 * NDIM);
  layernorm_kernel<<<N, 256, 0, stream>>>(out, ln2g, ln2b, out);
}